// GINConvNet_72146860638340
// MI455X (gfx1250) — compile-verified
//
#include <hip/hip_runtime.h>
#include <hip/hip_bf16.h>

typedef __attribute__((ext_vector_type(2))) float v2f;
typedef __attribute__((ext_vector_type(8))) float v8f;

static __device__ __forceinline__ v8f wmma_f32(v2f a, v2f b, v8f c) {
    // D = A(16x4 f32) x B(4x16 f32) + C(16x16 f32); wave32, EXEC all-ones.
    return __builtin_amdgcn_wmma_f32_16x16x4_f32(
        /*neg_a=*/false, a, /*neg_b=*/false, b,
        /*c_mod=*/(short)0, c, /*reuse_a=*/false, /*reuse_b=*/false);
}

static __device__ __forceinline__ int imin(int a, int b) { return a < b ? a : b; }

// ---------------------------------------------------------------------------
// Generic GEMM: C[M,N] = act(A[M,K] @ B[K,N] + bias[N]).
// One wave computes a 16x32 tile (two 16x16 subtiles sharing the A fragment):
// per K-step = 1 b64 A load + 4 b32 B loads + 2 WMMAs (~2 loads/WMMA).
// OOB rows/cols are index-clamped; their outputs are never stored.
// ---------------------------------------------------------------------------
__global__ __launch_bounds__(256) void gemm_wmma_kernel(
    const float* __restrict__ A, const float* __restrict__ B,
    const float* __restrict__ bias, float* __restrict__ C,
    int M, int N, int K, int ldc, int relu)
{
    int wave = blockIdx.x * (blockDim.x >> 5) + (threadIdx.x >> 5);
    int lane = threadIdx.x & 31;
    int ntiles = (N + 31) >> 5;                    // 32-wide super-tiles
    int mtiles = (M + 15) >> 4;
    if (wave >= mtiles * ntiles) return;           // uniform per wave
    int mt = wave / ntiles, nt = wave % ntiles;
    int m0 = mt << 4, n0 = nt << 5;
    int half = lane >> 4;                          // K-pair selector (0/1)
    int lidx = lane & 15;
    int am  = imin(m0 + lidx, M - 1);              // clamped A row
    int bn0 = imin(n0 + lidx,      N - 1);         // clamped B cols
    int bn1 = imin(n0 + 16 + lidx, N - 1);

    const float* ap  = A + (size_t)am * K + 2 * half;
    const float* bp0 = B + bn0 + (size_t)(2 * half) * N;
    const float* bp1 = B + bn1 + (size_t)(2 * half) * N;
    int K4 = K & ~3;

    v8f acc0 = {}, acc1 = {};
    #pragma unroll 4
    for (int k0 = 0; k0 < K4; k0 += 4) {
        v2f a = *(const v2f*)ap;                   // contiguous, 8B aligned
        v2f b0; b0.x = bp0[0]; b0.y = bp0[N];
        v2f b1; b1.x = bp1[0]; b1.y = bp1[N];
        acc0 = wmma_f32(a, b0, acc0);
        acc1 = wmma_f32(a, b1, acc1);
        ap += 4;
        bp0 += (size_t)4 * N;
        bp1 += (size_t)4 * N;
    }
    if (K4 < K) {                                  // K remainder (e.g. K=78)
        int kk = K4 + 2 * half;
        v2f a, b0, b1;
        a.x  = (kk     < K) ? ap[0]  : 0.0f;
        a.y  = (kk + 1 < K) ? ap[1]  : 0.0f;
        b0.x = (kk     < K) ? bp0[0] : 0.0f;
        b0.y = (kk + 1 < K) ? bp0[N] : 0.0f;
        b1.x = (kk     < K) ? bp1[0] : 0.0f;
        b1.y = (kk + 1 < K) ? bp1[N] : 0.0f;
        acc0 = wmma_f32(a, b0, acc0);              // full EXEC restored here
        acc1 = wmma_f32(a, b1, acc1);
    }

    #pragma unroll
    for (int t = 0; t < 2; ++t) {
        int col = n0 + 16 * t + lidx;
        if (col < N) {
            float bv = bias[col];
            v8f acc = t ? acc1 : acc0;
            #pragma unroll
            for (int r = 0; r < 8; ++r) {
                int row = m0 + r + 8 * half;
                if (row < M) {
                    float v = acc[r] + bv;
                    if (relu) v = fmaxf(v, 0.0f);
                    C[row * ldc + col] = v;
                }
            }
        }
    }
}

// ---------------------------------------------------------------------------
// Implicit-GEMM conv1d (window = 8) from an embedding table:
//   in[b,ci,l] = emb[tok[b,ci]][l], Lin = 256 (embedding width)
//   out[b,co,l] = relu( sum_{ci,k} in[b,ci,l+k] * W[co,ci,k] + bias[co] )
// GEMM view per b: M=Cout, N=Lout, K=Cin*8 (mult. of 8 -> no remainder).
// One wave = 16(co) x 32(l) tile: weight fragment AND token-row lookup are
// shared by both 16x16 subtiles. kk is even so kk,kk+1 share one ci group.
// ---------------------------------------------------------------------------
__global__ __launch_bounds__(256) void conv_embed_wmma_kernel(
    const int* __restrict__ tok, const float* __restrict__ emb,
    const float* __restrict__ W, const float* __restrict__ bias,
    float* __restrict__ out, int Batch, int Cin, int Cout, int Lout)
{
    int wave = blockIdx.x * (blockDim.x >> 5) + (threadIdx.x >> 5);
    int lane = threadIdx.x & 31;
    int ctiles = (Cout + 15) >> 4;
    int ltiles = (Lout + 31) >> 5;                 // 32-wide L super-tiles
    int per_b = ctiles * ltiles;
    if (wave >= Batch * per_b) return;
    int b   = wave / per_b;
    int rem = wave % per_b;
    int c0 = (rem / ltiles) << 4;
    int l0 = (rem % ltiles) << 5;
    int half = lane >> 4, lidx = lane & 15;
    int co   = imin(c0 + lidx, Cout - 1);          // clamped weight row
    int lc0  = imin(l0 + lidx,      Lout - 1);     // clamped output cols
    int lc1  = imin(l0 + 16 + lidx, Lout - 1);
    int K    = Cin * 8;
    const int* tokb = tok + b * Cin;
    const float* wp = W + (size_t)co * K + 2 * half;

    v8f acc0 = {}, acc1 = {};
    int kk = 2 * half;
    #pragma unroll 4
    for (int k0 = 0; k0 < K; k0 += 4) {
        v2f a = *(const v2f*)wp;                   // contiguous, 8B aligned
        int ci  = kk >> 3;
        int kof = kk & 7;                          // even, <= 6
        const float* er = emb + tokb[ci] * 256 + kof;   // one token fetch
        v2f b0; b0.x = er[lc0]; b0.y = er[lc0 + 1];
        v2f b1; b1.x = er[lc1]; b1.y = er[lc1 + 1];
        acc0 = wmma_f32(a, b0, acc0);
        acc1 = wmma_f32(a, b1, acc1);
        wp += 4;
        kk += 4;
    }
    #pragma unroll
    for (int t = 0; t < 2; ++t) {
        int l = l0 + 16 * t + lidx;
        if (l < Lout) {
            v8f acc = t ? acc1 : acc0;
            #pragma unroll
            for (int r = 0; r < 8; ++r) {
                int row = c0 + r + 8 * half;
                if (row < Cout)
                    out[(b * Cout + row) * Lout + l] = fmaxf(acc[r] + bias[row], 0.0f);
            }
        }
    }
}

// Same implicit-GEMM conv but input is a dense [B, Cin, Lin] buffer.
__global__ __launch_bounds__(256) void conv_wmma_kernel(
    const float* __restrict__ in, const float* __restrict__ W,
    const float* __restrict__ bias, float* __restrict__ out,
    int Batch, int Cin, int Lin, int Cout, int Lout)
{
    int wave = blockIdx.x * (blockDim.x >> 5) + (threadIdx.x >> 5);
    int lane = threadIdx.x & 31;
    int ctiles = (Cout + 15) >> 4;
    int ltiles = (Lout + 31) >> 5;
    int per_b = ctiles * ltiles;
    if (wave >= Batch * per_b) return;
    int b   = wave / per_b;
    int rem = wave % per_b;
    int c0 = (rem / ltiles) << 4;
    int l0 = (rem % ltiles) << 5;
    int half = lane >> 4, lidx = lane & 15;
    int co  = imin(c0 + lidx, Cout - 1);
    int lc0 = imin(l0 + lidx,      Lout - 1);
    int lc1 = imin(l0 + 16 + lidx, Lout - 1);
    int K   = Cin * 8;
    const float* inb = in + (size_t)b * Cin * Lin;
    const float* wp  = W + (size_t)co * K + 2 * half;

    v8f acc0 = {}, acc1 = {};
    int kk = 2 * half;
    #pragma unroll 4
    for (int k0 = 0; k0 < K; k0 += 4) {
        v2f a = *(const v2f*)wp;
        int ci  = kk >> 3;
        int kof = kk & 7;                          // even, <= 6
        const float* ir = inb + ci * Lin + kof;
        v2f b0; b0.x = ir[lc0]; b0.y = ir[lc0 + 1];
        v2f b1; b1.x = ir[lc1]; b1.y = ir[lc1 + 1];
        acc0 = wmma_f32(a, b0, acc0);
        acc1 = wmma_f32(a, b1, acc1);
        wp += 4;
        kk += 4;
    }
    #pragma unroll
    for (int t = 0; t < 2; ++t) {
        int l = l0 + 16 * t + lidx;
        if (l < Lout) {
            v8f acc = t ? acc1 : acc0;
            #pragma unroll
            for (int r = 0; r < 8; ++r) {
                int row = c0 + r + 8 * half;
                if (row < Cout)
                    out[(b * Cout + row) * Lout + l] = fmaxf(acc[r] + bias[row], 0.0f);
            }
        }
    }
}

// ---------------------------------------------------------------------------
// Elementwise / scatter kernels
// ---------------------------------------------------------------------------
__global__ void copy_kernel(float* __restrict__ dst, const float* __restrict__ src, int n) {
    int i = blockIdx.x * blockDim.x + threadIdx.x;
    if (i < n) dst[i] = src[i];
}

__global__ void zero_kernel(float* __restrict__ dst, int n) {
    int i = blockIdx.x * blockDim.x + threadIdx.x;
    if (i < n) dst[i] = 0.0f;
}

// t[dst[e], c] += h[src[e], c]  (t was pre-initialized with h, giving x + agg)
__global__ void edge_agg_kernel(float* __restrict__ t, const float* __restrict__ h,
                                const int* __restrict__ src, const int* __restrict__ dst,
                                int E, int dim) {
    int i = blockIdx.x * blockDim.x + threadIdx.x;
    if (i >= E * dim) return;
    int e = i / dim, c = i % dim;
    atomicAdd(&t[dst[e] * dim + c], h[src[e] * dim + c]);
}

// h = gamma * (relu(g2) - mean) * rsqrt(var + 1e-5) + beta   (dim == 32)
__global__ void bn_relu_kernel(const float* __restrict__ g2,
                               const float* __restrict__ gamma, const float* __restrict__ beta,
                               const float* __restrict__ mean,  const float* __restrict__ var,
                               float* __restrict__ h, int total) {
    int i = blockIdx.x * blockDim.x + threadIdx.x;
    if (i >= total) return;
    int c = i & 31;
    float v = fmaxf(g2[i], 0.0f);
    h[i] = gamma[c] * (v - mean[c]) * rsqrtf(var[c] + 1e-5f) + beta[c];
}

// xg[batch[n], c] += h[n, c]   (dim == 32)
__global__ void graph_pool_kernel(const float* __restrict__ h, const int* __restrict__ batch,
                                  float* __restrict__ xg, int total) {
    int i = blockIdx.x * blockDim.x + threadIdx.x;
    if (i >= total) return;
    int node = i >> 5, c = i & 31;
    atomicAdd(&xg[batch[node] * 32 + c], h[i]);
}

// maxpool kernel=3 stride=3 over last axis: in [BC, Lin] -> out [BC, P]
__global__ void maxpool3_kernel(const float* __restrict__ in, float* __restrict__ out,
                                int BC, int Lin, int P) {
    int i = blockIdx.x * blockDim.x + threadIdx.x;
    if (i >= BC * P) return;
    int bc = i / P, p = i % P;
    const float* q = in + (size_t)bc * Lin + 3 * p;
    out[i] = fmaxf(q[0], fmaxf(q[1], q[2]));
}

// dst[r, c] = src[r*sld + c]   (gather strided rows into contiguous block)
__global__ void copy2d_kernel(float* __restrict__ dst, const float* __restrict__ src,
                              int rows, int cols, int sld) {
    int i = blockIdx.x * blockDim.x + threadIdx.x;
    if (i >= rows * cols) return;
    int r = i / cols, c = i % cols;
    dst[i] = src[r * sld + c];
}

// out[i] = dot(y2[i, 0:K], w) + b0
__global__ void out_head_kernel(const float* __restrict__ y2, const float* __restrict__ w,
                                const float* __restrict__ b, float* __restrict__ out,
                                int B, int K) {
    int i = blockIdx.x * blockDim.x + threadIdx.x;
    if (i >= B) return;
    float s = 0.0f;
    for (int k = 0; k < K; ++k) s += y2[i * K + k] * w[k];
    out[i] = s + b[0];
}

// ---------------------------------------------------------------------------
// Host orchestration
// ---------------------------------------------------------------------------
static inline int ceil_div(int a, int b) { return (a + b - 1) / b; }

static void launch_gemm(const float* A, const float* B, const float* bias, float* C,
                        int M, int N, int K, int ldc, int relu, hipStream_t s) {
    int waves = ceil_div(M, 16) * ceil_div(N, 32);
    int blocks = ceil_div(waves, 8);                  // 8 waves / 256-thread block
    gemm_wmma_kernel<<<blocks, 256, 0, s>>>(A, B, bias, C, M, N, K, ldc, relu);
}

static void launch_conv_embed(const int* tok, const float* emb, const float* W,
                              const float* bias, float* out,
                              int Batch, int Cin, int Cout, int Lout, hipStream_t s) {
    int waves = Batch * ceil_div(Cout, 16) * ceil_div(Lout, 32);
    conv_embed_wmma_kernel<<<ceil_div(waves, 8), 256, 0, s>>>(tok, emb, W, bias, out,
                                                              Batch, Cin, Cout, Lout);
}

static void launch_conv(const float* in, const float* W, const float* bias, float* out,
                        int Batch, int Cin, int Lin, int Cout, int Lout, hipStream_t s) {
    int waves = Batch * ceil_div(Cout, 16) * ceil_div(Lout, 32);
    conv_wmma_kernel<<<ceil_div(waves, 8), 256, 0, s>>>(in, W, bias, out,
                                                        Batch, Cin, Lin, Cout, Lout);
}

static void launch_pool(const float* in, float* out, int BC, int Lin, int P, hipStream_t s) {
    maxpool3_kernel<<<ceil_div(BC * P, 256), 256, 0, s>>>(in, out, BC, Lin, P);
}

#define NNODES 20480
#define NEDGES 81920
#define NGRAPH 512

extern "C" void kernel_launch(void* const* d_in, const int* in_sizes, int n_in,
                              void* d_out, int out_size, void* d_ws, size_t ws_size,
                              hipStream_t stream)
{
    (void)in_sizes; (void)n_in; (void)out_size; (void)ws_size;

    const float* x      = (const float*)d_in[0];   // [20480, 78]
    const int*   edge   = (const int*)  d_in[1];   // [2, 81920]
    const int*   batch  = (const int*)  d_in[2];   // [20480]
    const int*   target = (const int*)  d_in[3];   // [512, 735]
    const int*   drug   = (const int*)  d_in[4];   // [512, 100]

    int p = 5;
    const float *gW1[5], *gb1[5], *gW2[5], *gb2[5], *gga[5], *gbe[5], *gme[5], *gva[5];
    for (int i = 0; i < 5; ++i) {
        gW1[i] = (const float*)d_in[p++]; gb1[i] = (const float*)d_in[p++];
        gW2[i] = (const float*)d_in[p++]; gb2[i] = (const float*)d_in[p++];
        gga[i] = (const float*)d_in[p++]; gbe[i] = (const float*)d_in[p++];
        gme[i] = (const float*)d_in[p++]; gva[i] = (const float*)d_in[p++];
    }
    const float* fc1_xd_W  = (const float*)d_in[p++];
    const float* fc1_xd_b  = (const float*)d_in[p++];
    const float* emb_xt    = (const float*)d_in[p++];
    const float* cxt1_W    = (const float*)d_in[p++];
    const float* cxt1_b    = (const float*)d_in[p++];
    const float* cxt2_W    = (const float*)d_in[p++];
    const float* cxt2_b    = (const float*)d_in[p++];
    const float* cxt3_W    = (const float*)d_in[p++];
    const float* cxt3_b    = (const float*)d_in[p++];
    const float* fc1_xt_W  = (const float*)d_in[p++];
    const float* fc1_xt_b  = (const float*)d_in[p++];
    const float* fc2_xt_W  = (const float*)d_in[p++];
    const float* fc2_xt_b  = (const float*)d_in[p++];
    const float* emb_xds   = (const float*)d_in[p++];
    const float* cxd1_W    = (const float*)d_in[p++];
    const float* cxd1_b    = (const float*)d_in[p++];
    const float* cxd2_W    = (const float*)d_in[p++];
    const float* cxd2_b    = (const float*)d_in[p++];
    const float* cxd3_W    = (const float*)d_in[p++];
    const float* cxd3_b    = (const float*)d_in[p++];
    const float* fc1_xds_W = (const float*)d_in[p++];
    const float* fc1_xds_b = (const float*)d_in[p++];
    const float* fc2_xds_W = (const float*)d_in[p++];
    const float* fc2_xds_b = (const float*)d_in[p++];
    const float* fc1_W     = (const float*)d_in[p++];
    const float* fc1_b     = (const float*)d_in[p++];
    const float* fc2_W     = (const float*)d_in[p++];
    const float* fc2_b     = (const float*)d_in[p++];
    const float* out_W     = (const float*)d_in[p++];
    const float* out_b     = (const float*)d_in[p++];

    // -------- workspace layout (floats) --------
    float* ws = (float*)d_ws;
    float* BUF_A = ws;                 // 4,080,000  (largest: xt conv1 out 512*32*249)
    float* BUF_B = ws + 4080000;       // 1,360,000  (pool outputs, g1)
    float* BUF_C = ws + 5440000;       // 656,000    (g2)
    float* BUF_H = ws + 6096000;       // 656,000    (GIN hidden, persists)
    float* XG    = ws + 6752000;       // 16,400     (pooled graph features pre-FC)
    float* XC    = ws + 6768400;       // 196,608    (concat [xg | xd | xt], ld=384)

    const int* esrc = edge;            // edge_index[0]
    const int* edst = edge + NEDGES;   // edge_index[1]

    // ---------------- GIN stack (5 layers) ----------------
    const float* hin = x;
    int dim = 78;
    for (int l = 0; l < 5; ++l) {
        int nelem = NNODES * dim;
        copy_kernel<<<ceil_div(nelem, 256), 256, 0, stream>>>(BUF_A, hin, nelem);
        edge_agg_kernel<<<ceil_div(NEDGES * dim, 256), 256, 0, stream>>>(
            BUF_A, hin, esrc, edst, NEDGES, dim);
        // g1 = relu(t @ W1 + b1)
        launch_gemm(BUF_A, gW1[l], gb1[l], BUF_B, NNODES, 32, dim, 32, 1, stream);
        // g2 = g1 @ W2 + b2
        launch_gemm(BUF_B, gW2[l], gb2[l], BUF_C, NNODES, 32, 32, 32, 0, stream);
        // h = bn(relu(g2))
        bn_relu_kernel<<<ceil_div(NNODES * 32, 256), 256, 0, stream>>>(
            BUF_C, gga[l], gbe[l], gme[l], gva[l], BUF_H, NNODES * 32);
        hin = BUF_H;
        dim = 32;
    }

    // ---------------- global_add_pool + fc1_xd ----------------
    zero_kernel<<<ceil_div(NGRAPH * 32, 256), 256, 0, stream>>>(XG, NGRAPH * 32);
    graph_pool_kernel<<<ceil_div(NNODES * 32, 256), 256, 0, stream>>>(
        BUF_H, batch, XG, NNODES * 32);
    // xg = relu(XG @ fc1_xd_W + b) -> XC[:, 0:128]
    launch_gemm(XG, fc1_xd_W, fc1_xd_b, XC, NGRAPH, 128, 32, 384, 1, stream);
    // second output: xg -> d_out[512 : 512+65536]
    copy2d_kernel<<<ceil_div(NGRAPH * 128, 256), 256, 0, stream>>>(
        (float*)d_out + NGRAPH, XC, NGRAPH, 128, 384);

    // ---------------- xt tower (protein CNN) ----------------
    launch_conv_embed(target, emb_xt, cxt1_W, cxt1_b, BUF_A, NGRAPH, 735, 32, 249, stream);
    launch_pool(BUF_A, BUF_B, NGRAPH * 32, 249, 83, stream);
    launch_conv(BUF_B, cxt2_W, cxt2_b, BUF_A, NGRAPH, 32, 83, 64, 76, stream);
    launch_pool(BUF_A, BUF_B, NGRAPH * 64, 76, 25, stream);
    launch_conv(BUF_B, cxt3_W, cxt3_b, BUF_A, NGRAPH, 64, 25, 128, 18, stream);
    launch_pool(BUF_A, BUF_B, NGRAPH * 128, 18, 6, stream);
    // flatten [512,768] -> fc1 (relu) -> fc2 into XC[:, 256:384]
    launch_gemm(BUF_B, fc1_xt_W, fc1_xt_b, BUF_A, NGRAPH, 1024, 768, 1024, 1, stream);
    launch_gemm(BUF_A, fc2_xt_W, fc2_xt_b, XC + 256, NGRAPH, 128, 1024, 384, 0, stream);

    // ---------------- xd tower (drug-sequence CNN) ----------------
    launch_conv_embed(drug, emb_xds, cxd1_W, cxd1_b, BUF_A, NGRAPH, 100, 25, 249, stream);
    launch_pool(BUF_A, BUF_B, NGRAPH * 25, 249, 83, stream);
    launch_conv(BUF_B, cxd2_W, cxd2_b, BUF_A, NGRAPH, 25, 83, 50, 76, stream);
    launch_pool(BUF_A, BUF_B, NGRAPH * 50, 76, 25, stream);
    launch_conv(BUF_B, cxd3_W, cxd3_b, BUF_A, NGRAPH, 50, 25, 100, 18, stream);
    launch_pool(BUF_A, BUF_B, NGRAPH * 100, 18, 6, stream);
    // flatten [512,600] -> fc1 (relu) -> fc2 into XC[:, 128:256]
    launch_gemm(BUF_B, fc1_xds_W, fc1_xds_b, BUF_A, NGRAPH, 1024, 600, 1024, 1, stream);
    launch_gemm(BUF_A, fc2_xds_W, fc2_xds_b, XC + 128, NGRAPH, 128, 1024, 384, 0, stream);

    // ---------------- combined head ----------------
    launch_gemm(XC, fc1_W, fc1_b, BUF_A, NGRAPH, 1024, 384, 1024, 1, stream);   // relu
    launch_gemm(BUF_A, fc2_W, fc2_b, BUF_B, NGRAPH, 256, 1024, 256, 1, stream); // relu
    out_head_kernel<<<ceil_div(NGRAPH, 256), 256, 0, stream>>>(
        BUF_B, out_W, out_b, (float*)d_out, NGRAPH, 256);
}